// WriteHead_83159156785503
// MI455X (gfx1250) — compile-verified
//
#include <hip/hip_runtime.h>
#include <math.h>

#define BB 32
#define NN 8192
#define WW 256
#define RR 4
#define CTRL 775          // 3*W + R + 3
#define EPSF 1e-8f

typedef float v2f __attribute__((ext_vector_type(2)));
typedef float v8f __attribute__((ext_vector_type(8)));
typedef unsigned int v4u __attribute__((ext_vector_type(4)));
typedef int v4i __attribute__((ext_vector_type(4)));
typedef int v8i __attribute__((ext_vector_type(8)));

// ---------------- workspace layout (float offsets) ----------------
#define OFF_LOGIT ((size_t)0)
#define OFF_PHI   (OFF_LOGIT + (size_t)BB * NN)
#define OFF_USC   (OFF_PHI   + (size_t)BB * NN)
#define OFF_ALLOC (OFF_USC   + (size_t)BB * NN)
#define OFF_ER    (OFF_ALLOC + (size_t)BB * NN)
#define OFF_ADD   (OFF_ER    + (size_t)BB * WW)
#define OFF_FG    (OFF_ADD   + (size_t)BB * WW)
#define OFF_KN    (OFF_FG    + (size_t)BB * RR)
#define OFF_BETA  (OFF_KN    + (size_t)BB)
#define OFF_AG    (OFF_BETA  + (size_t)BB)
#define OFF_WG    (OFF_AG    + (size_t)BB)
#define OFF_MAX   (OFF_WG    + (size_t)BB)
#define OFF_DEN   (OFF_MAX   + (size_t)BB)

__device__ __forceinline__ float sigm(float x) { return 1.0f / (1.0f + expf(-x)); }

// ================= kernel 1: parse controls =================
// 32 blocks x 256 threads
__global__ void k_controls(const float* __restrict__ ctrl, float* __restrict__ ws) {
    const int b = blockIdx.x;
    const int t = threadIdx.x;                 // 0..255 (== W)
    const float* c = ctrl + (size_t)b * CTRL;
    __shared__ float red[WW];

    float kv = c[t];
    red[t] = kv * kv;
    __syncthreads();
    for (int s = 128; s > 0; s >>= 1) {
        if (t < s) red[t] += red[t + s];
        __syncthreads();
    }

    ws[OFF_ER  + (size_t)b * WW + t] = sigm(c[WW + t]);       // erase_vector
    ws[OFF_ADD + (size_t)b * WW + t] = c[2 * WW + t];         // add_vector (aligned copy)
    if (t < RR) ws[OFF_FG + b * RR + t] = sigm(c[3 * WW + t]); // free gates
    if (t == 0) {
        float x = c[3 * WW + RR];                              // beta raw
        float sp = (x > 20.0f) ? x : log1pf(expf(x));          // softplus
        ws[OFF_KN   + b] = sqrtf(red[0]);                      // ||key||
        ws[OFF_BETA + b] = 1.0f + sp;
        ws[OFF_AG   + b] = sigm(c[3 * WW + RR + 1]);
        ws[OFF_WG   + b] = sigm(c[3 * WW + RR + 2]);
    }
}

// ================= kernel 2: phi and u_scaled =================
// B*N threads
__global__ void k_phi(const float* __restrict__ rw, const float* __restrict__ us,
                      const float* __restrict__ pww, float* __restrict__ ws) {
    const size_t gid = (size_t)blockIdx.x * blockDim.x + threadIdx.x; // < B*N
    const int b = (int)(gid >> 13);
    const int n = (int)(gid & (NN - 1));
    float p = 1.0f;
#pragma unroll
    for (int r = 0; r < RR; ++r) {
        float fg = ws[OFF_FG + b * RR + r];
        p *= (1.0f - fg * rw[((size_t)b * RR + r) * NN + n]);
    }
    float u0 = us[gid];
    float u = (u0 + pww[gid] * (1.0f - u0)) * p;
    ws[OFF_PHI + gid] = p;
    ws[OFF_USC + gid] = u * (1.0f - EPSF) + EPSF;
}

// ======= kernel 3: TDM-staged, WMMA f32 dots + row norms -> logits =======
// grid = B * (N/32) = 8192 blocks, 64 threads (2 waves), 32 rows per block.
__global__ void k_dots(const float* __restrict__ mem, const float* __restrict__ ctrl,
                       float* __restrict__ ws) {
    __shared__ float sTile[32 * WW];   // 32 rows x 256 f32 = 32 KB
    __shared__ float sKey[WW];
    __shared__ float sDot[32];

    const int b    = blockIdx.x >> 8;
    const int row0 = (blockIdx.x & 255) << 5;
    const int tid  = threadIdx.x;

    // stage key (4 floats per thread; controls rows are only 4B aligned -> scalar loads)
    const float* c = ctrl + (size_t)b * CTRL;
#pragma unroll
    for (int q = 0; q < 4; ++q) sKey[tid * 4 + q] = c[tid * 4 + q];

    // --- Tensor Data Mover: DMA 32x256 f32 tile into LDS (issued by wave 0) ---
    if (tid < 32) {
        unsigned long long gaddr =
            (unsigned long long)(const void*)(mem + ((size_t)b * NN + row0) * WW);
        unsigned int ldsAddr =
            (unsigned int)__builtin_amdgcn_readfirstlane((int)(size_t)(void*)sTile);
        unsigned int galo =
            (unsigned int)__builtin_amdgcn_readfirstlane((int)(unsigned int)(gaddr & 0xffffffffull));
        unsigned int gahi =
            (unsigned int)__builtin_amdgcn_readfirstlane((int)(unsigned int)(gaddr >> 32));

        // D# group 0: count=1 | lds_addr | global_addr[56:0] | type=2
        v4u g0 = { 1u, ldsAddr, galo, (gahi & 0x1ffffffu) | 0x80000000u };
        // D# group 1: data_size=4B; tensor_dim0=256; tensor_dim1=32;
        //             tile_dim0=256; tile_dim1=32; tensor_dim0_stride=256
        v8i g1 = { (int)0x00020000,          // data_size=2 (4 bytes)
                   (int)(WW << 16),          // tensor_dim0 low16 @ [31:16]
                   (int)(32 << 16),          // tensor_dim1 low16 @ [31:16]
                   (int)(WW << 16),          // tile_dim0 @ [31:16]
                   (int)32,                  // tile_dim1 @ [15:0]
                   (int)WW,                  // tensor_dim0_stride low32
                   0, 0 };
        v4i g2 = { 0, 0, 0, 0 };
        v4i g3 = { 0, 0, 0, 0 };
        v8i g4 = { 0, 0, 0, 0, 0, 0, 0, 0 };
        __builtin_amdgcn_tensor_load_to_lds(g0, g1, g2, g3, g4, 0);
    }
    __builtin_amdgcn_s_wait_tensorcnt(0);
    __syncthreads();

    // --- WMMA f32 16x16x4: 16 rows per wave, K-loop of 64 ---
    const int wave = tid >> 5;
    const int lane = tid & 31;
    const int m    = lane & 15;
    const int half = lane >> 4;
    const int r    = wave * 16 + m;            // row within 32-row tile

    v8f acc = { 0.f, 0.f, 0.f, 0.f, 0.f, 0.f, 0.f, 0.f };
#pragma unroll 4
    for (int k = 0; k < 64; ++k) {
        v2f a = *(const v2f*)(sTile + r * WW + 4 * k + 2 * half);  // A: 16x4 layout
        v2f kb = *(const v2f*)(sKey + 4 * k + 2 * half);           // B: key bcast cols
        acc = __builtin_amdgcn_wmma_f32_16x16x4_f32(
            false, a, false, kb, (short)0, acc, false, false);
    }
    // D: lane (half h) VGPR v holds dot for row v + 8*h, replicated over 16 cols
    if (m == 0) {
#pragma unroll
        for (int v = 0; v < 8; ++v) sDot[wave * 16 + half * 8 + v] = acc[v];
    }

    // --- row norms from the same LDS tile (each lane: half a row) ---
    float ss = 0.0f;
    const float4* rp = (const float4*)(sTile + r * WW + half * 128);
#pragma unroll 8
    for (int q = 0; q < 32; ++q) {
        float4 x = rp[q];
        ss += x.x * x.x + x.y * x.y + x.z * x.z + x.w * x.w;
    }
    ss += __shfl_xor(ss, 16, 32);
    __syncthreads();

    if (half == 0) {
        float dot = sDot[wave * 16 + m];
        float kn  = ws[OFF_KN + b];
        float bta = ws[OFF_BETA + b];
        float logit = dot / (kn * sqrtf(ss) + EPSF) * bta;
        ws[OFF_LOGIT + (size_t)b * NN + row0 + r] = logit;
    }
}

// ================= kernel 4: softmax stats per batch =================
// 32 blocks x 1024 threads
__global__ void k_softmax(float* __restrict__ ws) {
    const int b = blockIdx.x;
    const int t = threadIdx.x;
    __shared__ float red[1024];
    const float* lg = ws + OFF_LOGIT + (size_t)b * NN;

    float mx = -3.4e38f;
    for (int i = t; i < NN; i += 1024) mx = fmaxf(mx, lg[i]);
    red[t] = mx;
    __syncthreads();
    for (int s = 512; s > 0; s >>= 1) {
        if (t < s) red[t] = fmaxf(red[t], red[t + s]);
        __syncthreads();
    }
    const float M = red[0];
    __syncthreads();

    float sum = 0.0f;
    for (int i = t; i < NN; i += 1024) sum += expf(lg[i] - M);
    red[t] = sum;
    __syncthreads();
    for (int s = 512; s > 0; s >>= 1) {
        if (t < s) red[t] += red[t + s];
        __syncthreads();
    }
    if (t == 0) { ws[OFF_MAX + b] = M; ws[OFF_DEN + b] = red[0]; }
}

// ===== kernel 5: stable bitonic argsort + cumprod scan + alloc scatter =====
// 32 blocks x 1024 threads; LDS = 32K + 16K + 4K = 52 KB
__global__ void k_sort_alloc(float* __restrict__ ws) {
    const int b = blockIdx.x;
    const int t = threadIdx.x;
    __shared__ float su[NN];
    __shared__ unsigned short si[NN];
    __shared__ float tp[1024];

    const float* uin = ws + OFF_USC + (size_t)b * NN;
    for (int i = t; i < NN; i += 1024) { su[i] = uin[i]; si[i] = (unsigned short)i; }
    __syncthreads();

    // ascending bitonic sort, stable via (value, index) lexicographic compare
    for (int k = 2; k <= NN; k <<= 1) {
        for (int j = k >> 1; j > 0; j >>= 1) {
            for (int i = t; i < NN; i += 1024) {
                int ixj = i ^ j;
                if (ixj > i) {
                    float a = su[i], c = su[ixj];
                    unsigned short ia = si[i], ic = si[ixj];
                    bool agtc = (a > c) || (a == c && ia > ic);
                    bool up = ((i & k) == 0);
                    if (up ? agtc : !agtc) {
                        su[i] = c; su[ixj] = a; si[i] = ic; si[ixj] = ia;
                    }
                }
            }
            __syncthreads();
        }
    }

    // per-thread product of its 8 sorted values
    float p = 1.0f;
#pragma unroll
    for (int q = 0; q < 8; ++q) p *= su[t * 8 + q];
    tp[t] = p;
    __syncthreads();

    // inclusive multiplicative Hillis-Steele scan over 1024 thread-products
    for (int off = 1; off < 1024; off <<= 1) {
        float v = (t >= off) ? tp[t - off] : 1.0f;
        __syncthreads();
        tp[t] *= v;
        __syncthreads();
    }

    // exclusive prefix + scatter allocation scores
    float* alloc = ws + OFF_ALLOC + (size_t)b * NN;
    float running = (t == 0) ? 1.0f : tp[t - 1];
#pragma unroll
    for (int q = 0; q < 8; ++q) {
        int pos = t * 8 + q;
        float v = su[pos];
        alloc[si[pos]] = (1.0f - v) * running;   // (1 - u_i) * prod_{j<i} u_j
        running *= v;
    }
}

// ================= kernel 6: fused erase/write update =================
// B*N*W/4 threads, float4 path
__global__ void k_final(const float* __restrict__ mem, float* __restrict__ out,
                        const float* __restrict__ ws) {
    const size_t gid = (size_t)blockIdx.x * blockDim.x + threadIdx.x; // < B*N*64
    const int w4 = (int)(gid & 63);
    const size_t bn = gid >> 6;
    const int b = (int)(bn >> 13);

    float ag = ws[OFF_AG + b], wg = ws[OFF_WG + b];
    float cw = expf(ws[OFF_LOGIT + bn] - ws[OFF_MAX + b]) / ws[OFF_DEN + b];
    float ww = wg * (ag * ws[OFF_ALLOC + bn] + (1.0f - ag) * cw);
    float ph = ws[OFF_PHI + bn];

    float4 m  = ((const float4*)mem)[gid];
    float4 er = ((const float4*)(ws + OFF_ER))[(size_t)b * (WW / 4) + w4];
    float4 ad = ((const float4*)(ws + OFF_ADD))[(size_t)b * (WW / 4) + w4];

    float4 o;
    o.x = m.x * (1.0f - ww * er.x) * ph + ww * ad.x;
    o.y = m.y * (1.0f - ww * er.y) * ph + ww * ad.y;
    o.z = m.z * (1.0f - ww * er.z) * ph + ww * ad.z;
    o.w = m.w * (1.0f - ww * er.w) * ph + ww * ad.w;
    ((float4*)out)[gid] = o;
}

extern "C" void kernel_launch(void* const* d_in, const int* in_sizes, int n_in,
                              void* d_out, int out_size, void* d_ws, size_t ws_size,
                              hipStream_t stream) {
    const float* memory   = (const float*)d_in[0];
    const float* controls = (const float*)d_in[1];
    const float* rw       = (const float*)d_in[2];
    const float* usages   = (const float*)d_in[3];
    const float* pww      = (const float*)d_in[4];
    float* out = (float*)d_out;
    float* ws  = (float*)d_ws;

    k_controls<<<BB, WW, 0, stream>>>(controls, ws);
    k_phi<<<(BB * NN) / 256, 256, 0, stream>>>(rw, usages, pww, ws);
    k_dots<<<BB * (NN / 32), 64, 0, stream>>>(memory, controls, ws);
    k_softmax<<<BB, 1024, 0, stream>>>(ws);
    k_sort_alloc<<<BB, 1024, 0, stream>>>(ws);
    k_final<<<(size_t)BB * NN * (WW / 4) / 256, 256, 0, stream>>>(memory, out, ws);
}